// GraphGCNWithSequence_566935683609
// MI455X (gfx1250) — compile-verified
//
#include <hip/hip_runtime.h>
#include <hip/hip_bf16.h>
#include <stdint.h>

#define NN      50000
#define NE      1600000
#define FIN     16
#define HD      64
#define TT      8
#define NC      10

typedef __attribute__((ext_vector_type(2))) float v2f;
typedef __attribute__((ext_vector_type(8))) float v8f;

// LDS layout for W (K x 64): row-pair interleave so B-frag {row 2p+0, row 2p+1}
// at one column is a contiguous float2. Pair stride 160 floats == 32 banks (mod 64):
// lanes 0-15 (pair p) hit banks 0-31, lanes 16-31 (pair p+1) hit banks 32-63.
#define WPAIR_STRIDE 160
#define WLDS_FLOATS  (32 * WPAIR_STRIDE)   // up to K=64 -> 32 pairs (20 KB)

// ---------------- normalization precompute ----------------

__global__ void deg_init(float* deg, int n) {
    int i = blockIdx.x * blockDim.x + threadIdx.x;
    if (i < n) deg[i] = 1.0f;                       // self-loop weight = 1
}

__global__ void deg_scatter(const long long* __restrict__ dst,
                            const float* __restrict__ w,
                            float* __restrict__ deg, int e) {
    int i = blockIdx.x * blockDim.x + threadIdx.x;
    if (i < e) atomicAdd(&deg[(int)dst[i]], w[i]);
}

__global__ void dinv_kernel(float* d, int n) {
    int i = blockIdx.x * blockDim.x + threadIdx.x;
    if (i < n) {
        float v = d[i];
        d[i] = (v > 0.0f) ? rsqrtf(v) : 0.0f;
    }
}

__global__ void norm_kernel(const long long* __restrict__ src,
                            const long long* __restrict__ dst,
                            const float* __restrict__ w,
                            const float* __restrict__ dinv,
                            float* __restrict__ norm, int e) {
    int i = blockIdx.x * blockDim.x + threadIdx.x;
    if (i < e) norm[i] = dinv[(int)src[i]] * w[i] * dinv[(int)dst[i]];
}

// ---------------- WMMA fp32 GEMM:  C[n,64] (+)= A[n,K] @ W[K,64] ----------------
// W staged in LDS per block (shared by 4 waves); B-frags via ds_load_b64.
// One wave owns a 16-row block, 4 column tiles of 16, K stepped by 4
// (V_WMMA_F32_16X16X4_F32, full fp32 like the reference).

template <bool CONTIG>
__global__ void gemm_wmma_f32(const float* __restrict__ A, int aRowStride, int aEltStride,
                              const float* __restrict__ W, int K,
                              float* __restrict__ C, int accumulate, int n)
{
    __shared__ float ldsW[WLDS_FLOATS];

    // cooperative stage of W[K,64] into pair-interleaved LDS
    for (int idx = threadIdx.x; idx < K * HD; idx += blockDim.x) {
        int k = idx >> 6, col = idx & 63;
        ldsW[(k >> 1) * WPAIR_STRIDE + (col << 1) + (k & 1)] = W[idx];
    }
    __syncthreads();

    const int wave = threadIdx.x >> 5;
    const int lane = threadIdx.x & 31;
    const int mblk = blockIdx.x * (blockDim.x >> 5) + wave;
    const int m0   = mblk * 16;
    if (m0 >= n) return;                            // wave-uniform: EXEC all-ones below

    const int half = lane >> 4;                     // 0: lanes 0-15, 1: lanes 16-31
    const int l15  = lane & 15;

    v8f acc[4];
    #pragma unroll
    for (int j = 0; j < 4; ++j) acc[j] = (v8f){0.f,0.f,0.f,0.f,0.f,0.f,0.f,0.f};

    const float* arow = A + (unsigned)((m0 + l15) * aRowStride);

    for (int k0 = 0; k0 < K; k0 += 4) {
        // A frag (16x4): VGPR0 = K=k0 / k0+2 per half, VGPR1 = K=k0+1 / k0+3
        v2f a;
        if (CONTIG) {
            float2 t2 = *reinterpret_cast<const float2*>(arow + (unsigned)(k0 + 2 * half));
            a.x = t2.x; a.y = t2.y;
        } else {
            a.x = arow[(unsigned)((k0 + 2 * half + 0) * aEltStride)];
            a.y = arow[(unsigned)((k0 + 2 * half + 1) * aEltStride)];
        }

        const int pairBase = ((k0 >> 1) + half) * WPAIR_STRIDE + (l15 << 1);
        #pragma unroll
        for (int j = 0; j < 4; ++j) {               // 4 column tiles of 16
            float2 b2 = *reinterpret_cast<const float2*>(&ldsW[pairBase + (j << 5)]);
            v2f b; b.x = b2.x; b.y = b2.y;
            acc[j] = __builtin_amdgcn_wmma_f32_16x16x4_f32(
                false, a, false, b, (short)0, acc[j], false, false);
        }
    }

    // C/D layout: VGPR r -> row m0 + 8*half + r, col = j*16 + l15
    #pragma unroll
    for (int j = 0; j < 4; ++j) {
        float* crow = C + (unsigned)(m0 * HD + j * 16 + l15);
        #pragma unroll
        for (int r = 0; r < 8; ++r) {
            unsigned idx = (unsigned)((8 * half + r) * HD);
            if (accumulate) crow[idx] += acc[j][r];
            else            crow[idx]  = acc[j][r];
        }
    }
}

// ---------------- edge aggregation: agg[dst] += norm[e] * G[src] ----------------
// One thread per (edge, 4-feature chunk): float4 gather (L2-resident) + 4 f32 atomics.
// The 16 lanes of one edge load identical src/dst/norm -> single broadcast request.

__global__ void aggregate(const long long* __restrict__ src,
                          const long long* __restrict__ dst,
                          const float* __restrict__ norm,
                          const float* __restrict__ G,
                          float* __restrict__ agg, int e)
{
    long long tid = (long long)blockIdx.x * blockDim.x + threadIdx.x;
    int ei = (int)(tid >> 4);
    int c  = ((int)tid & 15) * 4;
    if (ei >= e) return;
    int s = (int)src[ei], d = (int)dst[ei];
    float nw = norm[ei];
    float4 v = *reinterpret_cast<const float4*>(G + (unsigned)(s * HD + c));
    float* o = agg + (unsigned)(d * HD + c);
    atomicAdd(o + 0, v.x * nw);
    atomicAdd(o + 1, v.y * nw);
    atomicAdd(o + 2, v.z * nw);
    atomicAdd(o + 3, v.w * nw);
}

// ---------------- epilogue: h = relu(agg + dinv^2 * G + b) (self-loop fused) ----------------

__global__ void conv_finalize(const float* __restrict__ agg,
                              const float* __restrict__ G,
                              const float* __restrict__ dinv,
                              const float* __restrict__ bias,
                              float* __restrict__ out, int n)
{
    int i = blockIdx.x * blockDim.x + threadIdx.x;
    if (i >= n * HD) return;
    int node = i >> 6, c = i & 63;
    float di = dinv[node];
    float v  = agg[i] + di * di * G[i] + bias[c];
    out[i] = fmaxf(v, 0.0f);
}

// ---------------- head: out = relu(acc + bseq) @ Wcls + bcls ----------------

__global__ void head_kernel(const float* __restrict__ acc,
                            const float* __restrict__ bseq,
                            const float* __restrict__ Wcls,
                            const float* __restrict__ bcls,
                            float* __restrict__ out, int n)
{
    int node = blockIdx.x * blockDim.x + threadIdx.x;
    if (node >= n) return;
    float o[NC];
    #pragma unroll
    for (int c = 0; c < NC; ++c) o[c] = bcls[c];
    const float* row = acc + (unsigned)(node * HD);
    #pragma unroll
    for (int h = 0; h < HD; ++h) {
        float v = fmaxf(row[h] + bseq[h], 0.0f);
        #pragma unroll
        for (int c = 0; c < NC; ++c) o[c] += v * Wcls[h * NC + c];
    }
    #pragma unroll
    for (int c = 0; c < NC; ++c) out[(unsigned)(node * NC + c)] = o[c];
}

// ---------------- driver ----------------

static inline size_t align256(size_t x) { return (x + 255) & ~(size_t)255; }

extern "C" void kernel_launch(void* const* d_in, const int* in_sizes, int n_in,
                              void* d_out, int out_size, void* d_ws, size_t ws_size,
                              hipStream_t stream) {
    const float*     x    = (const float*)d_in[0];      // [N, FIN, T]
    const long long* ei   = (const long long*)d_in[1];  // [2, E] int64
    const float*     ew   = (const float*)d_in[2];      // [E]
    const float*     W1   = (const float*)d_in[3];
    const float*     b1   = (const float*)d_in[4];
    const float*     W2   = (const float*)d_in[5];
    const float*     b2   = (const float*)d_in[6];
    const float*     W3   = (const float*)d_in[7];
    const float*     b3   = (const float*)d_in[8];
    const float*     Wseq = (const float*)d_in[9];      // [T*HD, HD]
    const float*     bseq = (const float*)d_in[10];
    const float*     Wcls = (const float*)d_in[11];     // [HD, NC]
    const float*     bcls = (const float*)d_in[12];
    float*           out  = (float*)d_out;

    const long long* src = ei;
    const long long* dst = ei + NE;

    // workspace carve (~58 MB)
    char* ws = (char*)d_ws;
    size_t off = 0;
    float* dinv   = (float*)(ws + off); off += align256((size_t)NN * 4);
    float* norm   = (float*)(ws + off); off += align256((size_t)NE * 4);
    float* bufG   = (float*)(ws + off); off += align256((size_t)NN * HD * 4);
    float* bufAgg = (float*)(ws + off); off += align256((size_t)NN * HD * 4);
    float* bufH   = (float*)(ws + off); off += align256((size_t)NN * HD * 4);
    float* accS   = (float*)(ws + off); off += align256((size_t)NN * HD * 4);
    (void)ws_size; (void)in_sizes; (void)n_in; (void)out_size;

    const size_t featBytes = (size_t)NN * HD * 4;

    // --- GCN normalization ---
    deg_init   <<<(NN + 255) / 256, 256, 0, stream>>>(dinv, NN);
    deg_scatter<<<(NE + 255) / 256, 256, 0, stream>>>(dst, ew, dinv, NE);
    dinv_kernel<<<(NN + 255) / 256, 256, 0, stream>>>(dinv, NN);
    norm_kernel<<<(NE + 255) / 256, 256, 0, stream>>>(src, dst, ew, dinv, norm, NE);

    hipMemsetAsync(accS, 0, featBytes, stream);

    const int rowBlocks = NN / 16;                           // 3125 (N divisible by 16)
    const dim3 gGemm((rowBlocks + 3) / 4), bGemm(128);       // 4 waves/block, 1 row-block/wave
    const long long aggThreads = (long long)NE * 16;
    const dim3 gAgg((unsigned)((aggThreads + 255) / 256)), bAgg(256);
    const dim3 gFin((NN * HD + 255) / 256), bFin(256);

    for (int t = 0; t < TT; ++t) {
        const float* xt = x + t;                             // elt stride T, row stride FIN*T

        // conv1: [N,16] @ W1  (strided read of x[:,:,t])
        gemm_wmma_f32<false><<<gGemm, bGemm, 0, stream>>>(xt, FIN * TT, TT, W1, FIN, bufG, 0, NN);
        hipMemsetAsync(bufAgg, 0, featBytes, stream);
        aggregate    <<<gAgg, bAgg, 0, stream>>>(src, dst, norm, bufG, bufAgg, NE);
        conv_finalize<<<gFin, bFin, 0, stream>>>(bufAgg, bufG, dinv, b1, bufH, NN);

        // conv2
        gemm_wmma_f32<true><<<gGemm, bGemm, 0, stream>>>(bufH, HD, 1, W2, HD, bufG, 0, NN);
        hipMemsetAsync(bufAgg, 0, featBytes, stream);
        aggregate    <<<gAgg, bAgg, 0, stream>>>(src, dst, norm, bufG, bufAgg, NE);
        conv_finalize<<<gFin, bFin, 0, stream>>>(bufAgg, bufG, dinv, b2, bufH, NN);

        // conv3
        gemm_wmma_f32<true><<<gGemm, bGemm, 0, stream>>>(bufH, HD, 1, W3, HD, bufG, 0, NN);
        hipMemsetAsync(bufAgg, 0, featBytes, stream);
        aggregate    <<<gAgg, bAgg, 0, stream>>>(src, dst, norm, bufG, bufAgg, NE);
        conv_finalize<<<gFin, bFin, 0, stream>>>(bufAgg, bufG, dinv, b3, bufH, NN);

        // sequence projection, accumulated per t: acc += h_t @ Wseq[t*64:(t+1)*64, :]
        gemm_wmma_f32<true><<<gGemm, bGemm, 0, stream>>>(bufH, HD, 1, Wseq + (size_t)t * HD * HD,
                                                         HD, accS, 1, NN);
    }

    head_kernel<<<(NN + 127) / 128, 128, 0, stream>>>(accS, bseq, Wcls, bcls, out, NN);
}